// PulseRetrievalLossFunction_49881750176369
// MI455X (gfx1250) — compile-verified
//
#include <hip/hip_runtime.h>
#include <math.h>
#include <stdint.h>

#define TSVAL 1.5e-15f
#define K2PI 6.28318530717958647692f

typedef float v2f __attribute__((ext_vector_type(2)));
typedef float v8f __attribute__((ext_vector_type(8)));

__device__ __forceinline__ float blockReduceSum(float v, float* red, int tid) {
  red[tid] = v; __syncthreads();
  #pragma unroll
  for (int off = 64; off > 0; off >>= 1) {
    if (tid < off) red[tid] = red[tid] + red[tid + off];
    __syncthreads();
  }
  float r = red[0]; __syncthreads();
  return r;
}

__device__ __forceinline__ float blockReduceMax(float v, float* red, int tid) {
  red[tid] = v; __syncthreads();
  #pragma unroll
  for (int off = 64; off > 0; off >>= 1) {
    if (tid < off) red[tid] = fmaxf(red[tid], red[tid + off]);
    __syncthreads();
  }
  float r = red[0]; __syncthreads();
  return r;
}

// Build twp[u] = e^{-2pi i u/1024} = (cos(2pi u/1024), -sin(2pi u/1024)), once per block.
__device__ __forceinline__ void buildTwiddle(float2* twp, int tid) {
  #pragma unroll
  for (int k = 0; k < 8; ++k) {
    int u = tid + (k << 7);
    float s, c;
    __sincosf(-K2PI * (float)u * (1.0f/1024.0f), &s, &c);
    twp[u] = make_float2(c, s);   // (cos, -sin)
  }
}

// 1024-pt complex DFT, 1024 = 16 x 64.
// Stage 1 (radix-16, stride 64) on V_WMMA_F32_16X16X4_F32; twiddles from LDS table.
// Stage 2: 16 DFT-64s as two DFT-8 passes (VALU, co-executes with matrix pipe).
// g2r/g2i may alias mr/mi (m fully consumed before g2 written; barriers in between).
__device__ __forceinline__ void fft1024(const float* mr, const float* mi,
                                        float* hr, float* hi,
                                        float* g2r, float* g2i,
                                        const float2* twp,
                                        int tid, float* outR, float* outI, int* outK)
{
  const int lane = tid & 31;
  const int wv   = tid >> 5;          // 16-column chunk of n2 for this wave
  const int rA   = lane & 15;         // A-row (k1) / B,D column within chunk
  const int kh   = lane >> 4;
  const int col  = (wv << 4) + rA;    // n2 column for B/D

  v8f accR = {0.f,0.f,0.f,0.f,0.f,0.f,0.f,0.f};
  v8f accI = {0.f,0.f,0.f,0.f,0.f,0.f,0.f,0.f};
  #pragma unroll
  for (int q = 0; q < 4; ++q) {       // K chunks of 4 over n1
    const int kx = 4*q + 2*kh;        // A layout: .x -> K=4q+2*half, .y -> K+1
    const int ky = kx + 1;
    float2 wx = twp[((rA*kx) & 15) << 6];   // e^{-2pi i (rA*kx)/16}
    float2 wy = twp[((rA*ky) & 15) << 6];
    v2f aC = { wx.x,  wy.x};          // cos
    v2f aS = {-wx.y, -wy.y};          // +sin
    v2f aN = { wx.y,  wy.y};          // -sin
    v2f bR = {mr[(kx<<6) + col], mr[(ky<<6) + col]};
    v2f bI = {mi[(kx<<6) + col], mi[(ky<<6) + col]};
    // Gr = cos*mr + sin*mi ; Gi = cos*mi - sin*mr
    accR = __builtin_amdgcn_wmma_f32_16x16x4_f32(false, aC, false, bR, (short)0, accR, false, false);
    accR = __builtin_amdgcn_wmma_f32_16x16x4_f32(false, aS, false, bI, (short)0, accR, false, false);
    accI = __builtin_amdgcn_wmma_f32_16x16x4_f32(false, aC, false, bI, (short)0, accI, false, false);
    accI = __builtin_amdgcn_wmma_f32_16x16x4_f32(false, aN, false, bR, (short)0, accI, false, false);
  }
  // twiddle e^{-2pi i r*col/1024} + scatter H. D layout: VGPR g -> M = g + 8*kh, N = rA.
  #pragma unroll
  for (int g = 0; g < 8; ++g) {
    const int r = g + (kh << 3);      // k1
    float gr = accR[g], gi = accI[g];
    float2 w = twp[r * col];          // r*col <= 945
    hr[(r<<6) + col] = gr*w.x - gi*w.y;
    hi[(r<<6) + col] = gr*w.y + gi*w.x;
  }
  __syncthreads();

  // Stage 2a: DFT-8 over a for b=t2, twiddle e^{-2pi i t2*ka/64} = twp[16*t2*ka]
  const float C8[8] = {1.f, 0.70710678f, 0.f, -0.70710678f, -1.f, -0.70710678f, 0.f, 0.70710678f};
  const float S8[8] = {0.f, -0.70710678f, -1.f, -0.70710678f, 0.f, 0.70710678f, 1.f, 0.70710678f};
  const int r2 = tid >> 3;            // k1 row: 0..15
  const int t2 = tid & 7;             // b in pass 1, ka in pass 2
  float xr[8], xi[8];
  #pragma unroll
  for (int a = 0; a < 8; ++a) {
    xr[a] = hr[(r2<<6) + (a<<3) + t2];
    xi[a] = hi[(r2<<6) + (a<<3) + t2];
  }
  #pragma unroll
  for (int ka = 0; ka < 8; ++ka) {
    float gr = 0.f, gi = 0.f;
    #pragma unroll
    for (int a = 0; a < 8; ++a) {
      const int u = (a*ka) & 7;
      gr += xr[a]*C8[u] - xi[a]*S8[u];
      gi += xr[a]*S8[u] + xi[a]*C8[u];
    }
    float2 w = twp[(t2*ka) << 4];
    g2r[(r2<<6) + (ka<<3) + t2] = gr*w.x - gi*w.y;
    g2i[(r2<<6) + (ka<<3) + t2] = gr*w.y + gi*w.x;
  }
  __syncthreads();
  // Stage 2b: DFT-8 over b for ka = t2; global k = r2 + 16*t2 + 128*kb
  #pragma unroll
  for (int kb = 0; kb < 8; ++kb) {
    float sr = 0.f, si = 0.f;
    #pragma unroll
    for (int b = 0; b < 8; ++b) {
      const int u = (b*kb) & 7;
      float vr = g2r[(r2<<6) + (t2<<3) + b];
      float vi = g2i[(r2<<6) + (t2<<3) + b];
      sr += vr*C8[u] - vi*S8[u];
      si += vr*S8[u] + vi*C8[u];
    }
    outR[kb] = sr; outI[kb] = si;
    outK[kb] = r2 + (t2<<4) + (kb<<7);
  }
}

// CDNA5 async global->LDS copy of one 4KB spectrogram row (two B128 per thread).
// GVS mode: SGPR64 base + per-lane byte offset; IOFFSET applies to both LDS and
// global addresses, so one address pair covers both 16B halves.
__device__ __forceinline__ void asyncLoadRow(const float* __restrict__ gbase,
                                             size_t row_byte_off,
                                             float* ldsdst, int tid) {
  uint32_t lds0 = (uint32_t)(uintptr_t)(ldsdst + tid*8);       // low 32b = LDS offset
  uint32_t v0   = (uint32_t)(row_byte_off + (size_t)tid*32u);
  asm volatile("global_load_async_to_lds_b128 %0, %1, %2\n\t"
               "global_load_async_to_lds_b128 %0, %1, %2 offset:16"
               :: "v"(lds0), "v"(v0), "s"(gbase) : "memory");
}

__device__ __forceinline__ void waitAsyncLE1() {
  asm volatile("s_wait_asynccnt 0x1" ::: "memory");
}

__global__ void init_kernel(float* __restrict__ p, int n) {
  int i = blockIdx.x * 256 + threadIdx.x;
  if (i < n) p[i] = 0.f;
}

// Center frequency: wC = sum(fftfreq(k) * |FFT(yta)[k]|^2) / sum(|FFT|^2)
__global__ __launch_bounds__(128) void cf_kernel(const float* __restrict__ pred,
                                                 float* __restrict__ wc) {
  __shared__ float mr[1024], mi[1024], hr[1024], hi[1024], red[128];
  __shared__ float2 twp[1024];
  const int s = blockIdx.x, tid = threadIdx.x;
  buildTwiddle(twp, tid);
  #pragma unroll
  for (int u = 0; u < 8; ++u) {
    int p = tid + (u<<7);
    mr[p] = pred[s*2048 + p];
    mi[p] = pred[s*2048 + 1024 + p];
  }
  __syncthreads();
  float oR[8], oI[8]; int oK[8];
  fft1024(mr, mi, hr, hi, mr, mi, twp, tid, oR, oI, oK);
  float wn = 0.f, wd = 0.f;
  #pragma unroll
  for (int kb = 0; kb < 8; ++kb) {
    float ps = oR[kb]*oR[kb] + oI[kb]*oI[kb];
    int k = oK[kb];
    float f = (float)(k < 512 ? k : k - 1024) * (1.0f/1024.0f);
    wn += f * ps; wd += ps;
  }
  wn = blockReduceSum(wn, red, tid);
  wd = blockReduceSum(wd, red, tid);
  if (tid == 0) wc[s] = wn / wd;
}

// One block = one sample x 16 delay rows; spectrogram rows async-prefetched to LDS.
__global__ __launch_bounds__(128) void frog_kernel(const float* __restrict__ pred,
                                                   const float* __restrict__ spec,
                                                   const float* __restrict__ wcv,
                                                   float* __restrict__ rowsum,
                                                   float* __restrict__ colR,
                                                   float* __restrict__ colM,
                                                   float* __restrict__ acc4) {
  __shared__ float yr[1024], yi[1024];
  __shared__ float mr[1024], mi[1024];     // also g2 scratch inside fft
  __shared__ float hr[1024], hi[1024];
  __shared__ float sp[2][1024];            // double-buffered spectrogram row
  __shared__ float csR[1024], csM[1024];
  __shared__ float red[128];
  __shared__ float2 twp[1024];
  const int s = blockIdx.y, dt = blockIdx.x, tid = threadIdx.x;
  buildTwiddle(twp, tid);
  #pragma unroll
  for (int u = 0; u < 8; ++u) {
    int p = tid + (u<<7);
    yr[p] = pred[s*2048 + p];
    yi[p] = pred[s*2048 + 1024 + p];
    csR[p] = 0.f; csM[p] = 0.f;
  }
  const float wC = wcv[s];
  const float shiftk = -2.0f * TSVAL * wC;   // arg of exp(-i*2*TS*wC*(p-512))
  float shc[8], shs[8];                      // shiftf is delay-row invariant: hoist
  #pragma unroll
  for (int u = 0; u < 8; ++u)
    __sincosf(shiftk * (float)(tid + (u<<7) - 512), &shs[u], &shc[u]);

  const size_t sbase = ((size_t)s << 20);    // s*1024*1024 elements
  // prefetch row 0
  asyncLoadRow(spec, (sbase + ((size_t)(dt<<4) << 10)) * 4u, sp[0], tid);

  float aSmr = 0.f, aSrr = 0.f, aSmm = 0.f, aMax = 0.f;
  __syncthreads();

  for (int i = 0; i < 16; ++i) {
    const int d = (dt<<4) + i;
    const int delay = d - 512;
    // issue prefetch of next row into the other buffer
    const int dn = (dt<<4) + (i < 15 ? i + 1 : 15);
    asyncLoadRow(spec, (sbase + ((size_t)dn << 10)) * 4u, sp[(i+1)&1], tid);
    #pragma unroll
    for (int u = 0; u < 8; ++u) {
      int p  = tid + (u<<7);
      int p2 = (p - delay) & 1023;               // (pos - delay) mod N
      float ar = yr[p], ai = yi[p];
      float br = yr[p2], bi = yi[p2];
      float pr  = ar*br - ai*bi;                 // yta[p] * rolled
      float pim = ar*bi + ai*br;
      mr[p] = pr*shc[u] - pim*shs[u];
      mi[p] = pr*shs[u] + pim*shc[u];
    }
    waitAsyncLE1();       // row i landed (in-order); row i+1 may still fly
    __syncthreads();
    float oR[8], oI[8]; int oK[8];
    fft1024(mr, mi, hr, hi, mr, mi, twp, tid, oR, oI, oK);
    const float* spc = sp[i & 1];
    float rsum = 0.f;
    #pragma unroll
    for (int kb = 0; kb < 8; ++kb) {
      float mag = TSVAL * sqrtf(oR[kb]*oR[kb] + oI[kb]*oI[kb]);   // shg[d][j]
      int j = (oK[kb] + 512) & 1023;             // output fftshift remap
      float sv = spc[j];
      rsum += mag;
      csR[j] += mag;                              // each j owned by exactly one thread
      csM[j] += sv;
      aSmr += mag * sv;
      aSrr += mag * mag;
      aSmm += sv * sv;
      aMax = fmaxf(aMax, sv);
    }
    float rtot = blockReduceSum(rsum, red, tid);  // barriers also protect LDS reuse
    if (tid == 0) rowsum[(s<<10) + d] = rtot;
    __syncthreads();
  }

  #pragma unroll
  for (int u = 0; u < 8; ++u) {
    int p = tid + (u<<7);
    atomicAdd(&colR[(s<<10) + p], csR[p]);
    atomicAdd(&colM[(s<<10) + p], csM[p]);
  }
  float v;
  v = blockReduceSum(aSmr, red, tid); if (tid == 0) atomicAdd(&acc4[s], v);
  v = blockReduceSum(aSrr, red, tid); if (tid == 0) atomicAdd(&acc4[32 + s], v);
  v = blockReduceSum(aSmm, red, tid); if (tid == 0) atomicAdd(&acc4[64 + s], v);
  v = blockReduceMax(aMax, red, tid);
  if (tid == 0) atomicMax((int*)&acc4[96 + s], __float_as_int(v));  // spec >= 0
}

__global__ void finalize_kernel(const float* __restrict__ rowsum,
                                const float* __restrict__ colR,
                                const float* __restrict__ colM,
                                const float* __restrict__ acc4,
                                float* __restrict__ out) {
  __shared__ float errs[32];
  const int s = threadIdx.x;   // 32 threads, one per sample
  float num = 0.f, den = 0.f;
  for (int k = 0; k < 1024; ++k) {
    float rs = rowsum[(s<<10) + k];
    num = fmaf(colM[(s<<10) + k], rs, num);  // sum(Tmeas @ Tref)
    den = fmaf(colR[(s<<10) + k], rs, den);  // sum(Tref @ Tref)
  }
  float mu  = num / den;
  float smr = acc4[s], srr = acc4[32+s], smm = acc4[64+s], mx = acc4[96+s];
  float r = smm - 2.f*mu*smr + mu*mu*srr;
  errs[s] = sqrtf(r / (1048576.f * mx * mx));
  __syncthreads();
  if (s == 0) {
    float a = 0.f;
    for (int t = 0; t < 32; ++t) a += errs[t];
    out[0] = a * (1.0f/32.0f);
  }
}

extern "C" void kernel_launch(void* const* d_in, const int* in_sizes, int n_in,
                              void* d_out, int out_size, void* d_ws, size_t ws_size,
                              hipStream_t stream) {
  const float* pred = (const float*)d_in[0];
  // d_in[1] = label: unused by the reference
  const float* spec = (const float*)d_in[2];

  float* ws     = (float*)d_ws;
  float* wc     = ws;                   // 32
  float* rowsum = ws + 32;              // 32*1024
  float* colR   = rowsum + 32*1024;     // 32*1024
  float* colM   = colR + 32*1024;       // 32*1024
  float* acc4   = colM + 32*1024;       // 128: [Smr|Srr|Smm|max] x 32

  const int zn = 3*32*1024 + 128;       // zero all accumulators every call
  hipLaunchKernelGGL(init_kernel, dim3((zn + 255)/256), dim3(256), 0, stream, rowsum, zn);
  hipLaunchKernelGGL(cf_kernel, dim3(32), dim3(128), 0, stream, pred, wc);
  hipLaunchKernelGGL(frog_kernel, dim3(64, 32), dim3(128), 0, stream,
                     pred, spec, wc, rowsum, colR, colM, acc4);
  hipLaunchKernelGGL(finalize_kernel, dim3(1), dim3(32), 0, stream,
                     rowsum, colR, colM, acc4, (float*)d_out);
}